// DifferentiableMemory_30468497998113
// MI455X (gfx1250) — compile-verified
//
#include <hip/hip_runtime.h>
#include <stdint.h>

typedef __attribute__((ext_vector_type(2))) float v2f;
typedef __attribute__((ext_vector_type(8))) float v8f;

#define CAND_CAP 8192

__device__ __forceinline__ unsigned ordkey(float f) {
    unsigned u = __float_as_uint(f);
    return (u & 0x80000000u) ? ~u : (u | 0x80000000u);
}

// ---------------------------------------------------------------------------
// Kernel 1: scores = mat @ query using V_WMMA_F32_16X16X4_F32.
// Block = 256 threads = 8 waves; each wave computes 16 rows.
// A (16x4 f32) per ISA layout: lanes 0-15 hold K=0..1 (VGPR0..1),
// lanes 16-31 hold K=2..3 -> contiguous v2f load at k0 + 2*half.
// B (4x16) replicates query across all N columns (VGPR j: rows j and j+2),
// so every column of D holds the dot product; we read column N=0
// (lane 0 -> M=0..7 in acc[0..7], lane 16 -> M=8..15).
// ---------------------------------------------------------------------------
__global__ void gemv_wmma(const float* __restrict__ mat,
                          const float* __restrict__ query,
                          float* __restrict__ scores,
                          int N, int D) {
    __shared__ float q[1024];
    for (int i = threadIdx.x; i < D; i += blockDim.x) q[i] = query[i];
    __syncthreads();

    const int wave = threadIdx.x >> 5;
    const int lane = threadIdx.x & 31;
    const int half = lane >> 4;     // 0: lanes 0-15, 1: lanes 16-31
    const int m    = lane & 15;     // A-matrix row within tile
    const int row_base = (blockIdx.x * 8 + wave) * 16;
    if (row_base >= N) return;

    if (row_base + 16 <= N) {
        const float* rowp = mat + (size_t)(row_base + m) * D;
        v8f acc = {};
        for (int k0 = 0; k0 < D; k0 += 4) {
            const int off = k0 + 2 * half;
            v2f a = *(const v2f*)(rowp + off);
            v2f b;
            b.x = q[off];
            b.y = q[off + 1];
            // 8 args: (neg_a, A, neg_b, B, c_mod, C, reuse_a, reuse_b)
            acc = __builtin_amdgcn_wmma_f32_16x16x4_f32(
                false, a, false, b, (short)0, acc, false, false);
        }
        if (lane == 0) {
            #pragma unroll
            for (int j = 0; j < 8; ++j) scores[row_base + j] = acc[j];
        } else if (lane == 16) {
            #pragma unroll
            for (int j = 0; j < 8; ++j) scores[row_base + 8 + j] = acc[j];
        }
    } else {
        // partial tail tile: plain scalar dot products (wave-uniform branch)
        int r = row_base + lane;
        if (lane < 16 && r < N) {
            const float* rowp = mat + (size_t)r * D;
            float s = 0.f;
            for (int k = 0; k < D; ++k) s += rowp[k] * q[k];
            scores[r] = s;
        }
    }
}

// ---------------------------------------------------------------------------
// Kernel 2: 2048-bin histogram of order-preserving float keys (top 11 bits)
// ---------------------------------------------------------------------------
__global__ void hist_kernel(const float* __restrict__ scores,
                            unsigned* __restrict__ hist, int N) {
    int stride = gridDim.x * blockDim.x;
    for (int i = blockIdx.x * blockDim.x + threadIdx.x; i < N; i += stride)
        atomicAdd(&hist[ordkey(scores[i]) >> 21], 1u);
}

// ---------------------------------------------------------------------------
// Kernel 3: find threshold bin containing the K-th largest score
// ---------------------------------------------------------------------------
__global__ void scan_kernel(const unsigned* __restrict__ hist,
                            unsigned* __restrict__ thresh, int K) {
    if (threadIdx.x == 0 && blockIdx.x == 0) {
        unsigned cum = 0;
        int t = 0;
        for (int b = 2047; b >= 0; --b) {
            cum += hist[b];
            if (cum >= (unsigned)K) { t = b; break; }
        }
        *thresh = (unsigned)t;
    }
}

// ---------------------------------------------------------------------------
// Kernel 4: compact candidates (bin >= threshold) into ws
// ---------------------------------------------------------------------------
__global__ void compact_kernel(const float* __restrict__ scores,
                               const unsigned* __restrict__ thresh,
                               unsigned* __restrict__ cnt,
                               unsigned* __restrict__ ck,
                               int* __restrict__ ci, int N) {
    unsigned T = *thresh;
    int stride = gridDim.x * blockDim.x;
    for (int i = blockIdx.x * blockDim.x + threadIdx.x; i < N; i += stride) {
        unsigned key = ordkey(scores[i]);
        if ((key >> 21) >= T) {
            unsigned pos = atomicAdd(cnt, 1u);
            if (pos < CAND_CAP) { ck[pos] = key; ci[pos] = i; }
        }
    }
}

// ---------------------------------------------------------------------------
// Kernel 5: 64 rounds of block-max over (key desc, index asc), then gather.
// Deterministic regardless of compaction order: full order on (key, ~idx).
// ---------------------------------------------------------------------------
__global__ void select_kernel(const float* __restrict__ mat,
                              unsigned* __restrict__ ck,
                              const int* __restrict__ ci,
                              const unsigned* __restrict__ cnt,
                              float* __restrict__ out,
                              int* __restrict__ idx_out,
                              int N, int D, int K) {
    __shared__ unsigned long long red[256];
    const int tid = threadIdx.x;
    unsigned count = *cnt;
    if (count > CAND_CAP) count = CAND_CAP;

    for (int r = 0; r < K; ++r) {
        unsigned long long best = 0;
        for (unsigned j = tid; j < count; j += blockDim.x) {
            if (ck[j] != 0) {
                unsigned long long p =
                    ((unsigned long long)ck[j] << 32) | (unsigned)(~ci[j]);
                if (p > best) best = p;
            }
        }
        red[tid] = best;
        __syncthreads();
        for (int s = 128; s > 0; s >>= 1) {
            if (tid < s && red[tid + s] > red[tid]) red[tid] = red[tid + s];
            __syncthreads();
        }
        unsigned long long win = red[0];
        __syncthreads();

        int widx = (int)(~(unsigned)(win & 0xFFFFFFFFu));
        if (win == 0) widx = 0;  // degenerate: fewer candidates than K

        // remove the winner (indices are unique -> exactly one slot)
        for (unsigned j = tid; j < count; j += blockDim.x)
            if (ci[j] == widx) ck[j] = 0;
        if (tid == 0) idx_out[r] = widx;

        // gather the winning row (coalesced)
        const float* rowp = mat + (size_t)widx * D;
        for (int c = tid; c < D; c += blockDim.x)
            out[(size_t)r * D + c] = rowp[c];
        __syncthreads();
    }
}

// ---------------------------------------------------------------------------
extern "C" void kernel_launch(void* const* d_in, const int* in_sizes, int n_in,
                              void* d_out, int out_size, void* d_ws, size_t ws_size,
                              hipStream_t stream) {
    const float* query = (const float*)d_in[0];
    const float* mat   = (const float*)d_in[1];
    const int D = in_sizes[0];
    const int N = in_sizes[1] / D;
    const int K = out_size / (D + 1);   // out = K*D floats + K indices

    // workspace layout
    float*    scores = (float*)d_ws;
    size_t    Npad   = ((size_t)N + 255) & ~(size_t)255;
    unsigned* hist   = (unsigned*)(scores + Npad);
    unsigned* cnt    = hist + 2048;
    unsigned* thresh = cnt + 1;
    unsigned* ck     = (unsigned*)((((uintptr_t)(thresh + 1)) + 255) & ~(uintptr_t)255);
    int*      ci     = (int*)(ck + CAND_CAP);

    hipMemsetAsync(hist, 0, (2048 + 2) * sizeof(unsigned), stream);

    const int gemv_blocks = (N + 127) / 128;   // 8 waves/block * 16 rows/wave
    gemv_wmma<<<gemv_blocks, 256, 0, stream>>>(mat, query, scores, N, D);
    hist_kernel<<<1024, 256, 0, stream>>>(scores, hist, N);
    scan_kernel<<<1, 32, 0, stream>>>(hist, thresh, K);
    compact_kernel<<<1024, 256, 0, stream>>>(scores, thresh, cnt, ck, ci, N);

    float* out     = (float*)d_out;
    int*   idx_out = (int*)(out + (size_t)K * D);
    select_kernel<<<1, 256, 0, stream>>>(mat, ck, ci, cnt, out, idx_out, N, D, K);
}